// TopK_39316130628142
// MI455X (gfx1250) — compile-verified
//
#include <hip/hip_runtime.h>

#define L_DIM   32768
#define K_SEL   64u
#define NTHR    512
#define PT      (L_DIM / NTHR)   /* 64 elements per thread   */
#define PT4     (PT / 4)         /* 16 float4 per thread     */
#define NB_MAX  4096
#define EQCAP   1024

// Match the builtin prototypes exactly: gcc-style v4i pointers in AS1 (global) / AS3 (LDS)
typedef int v4i __attribute__((vector_size(16)));
typedef __attribute__((address_space(1))) v4i* as1_v4i;
typedef __attribute__((address_space(3))) v4i* as3_v4i;

// Monotonic map: float bits -> unsigned key, order-preserving (descending float == descending key)
__device__ __forceinline__ unsigned f2key(unsigned u) {
  unsigned mask = (unsigned)((int)u >> 31) | 0x80000000u; // neg -> 0xFFFFFFFF, pos -> 0x80000000
  return u ^ mask;
}

__device__ __forceinline__ void async_g2l_b128(const float* g, float* l) {
  __builtin_amdgcn_global_load_async_to_lds_b128(
      (as1_v4i)(g), (as3_v4i)(l), 0, 0);
}

__device__ __forceinline__ void async_l2g_b128(float* g, float* l) {
  __builtin_amdgcn_global_store_async_from_lds_b128(
      (as1_v4i)(g), (as3_v4i)(l), 0, 0);
}

__global__ __launch_bounds__(NTHR)
void topk_relu_scatter_kernel(const float* __restrict__ x, float* __restrict__ out) {
  __shared__ __align__(16) float s_vals[L_DIM];   // 128 KB: full row staged in LDS
  __shared__ unsigned s_hist[NB_MAX];             // 16 KB
  __shared__ unsigned s_suf[NTHR];                // 2 KB suffix-scan workspace
  __shared__ unsigned s_bitmap[L_DIM / 32];       // 4 KB tie-selection bitmap
  __shared__ unsigned s_eqlist[EQCAP];            // 4 KB
  __shared__ unsigned s_eqcount;
  __shared__ unsigned s_selbin;
  __shared__ unsigned s_krem;

  const int tid = threadIdx.x;
  const size_t rowBase = (size_t)blockIdx.x * (size_t)L_DIM;
  const float* __restrict__ row  = x   + rowBase;
  float* __restrict__       orow = out + rowBase;

  // ---- Stage row into LDS via async DMA (ASYNCcnt path); overlap bitmap init ----
  #pragma unroll
  for (int i = 0; i < PT4; ++i) {
    int e4 = i * NTHR + tid;                      // coalesced: wave covers contiguous 512B
    async_g2l_b128(row + 4 * e4, &s_vals[4 * e4]);
  }
  for (int i = tid; i < L_DIM / 32; i += NTHR) s_bitmap[i] = 0;
  if (tid == 0) s_eqcount = 0;
  __builtin_amdgcn_s_wait_asynccnt(0);
  __syncthreads();

  // ---- Radix-select the exact K-th largest key: passes of 12,12,8 bits ----
  unsigned krem = K_SEL;
  unsigned prefHigh = 0;

  #pragma unroll
  for (int p = 0; p < 3; ++p) {
    const int shift = (p == 0) ? 20 : (p == 1) ? 8 : 0;
    const int bits  = (p == 2) ? 8 : 12;
    const unsigned nb = 1u << bits;

    for (unsigned i = tid; i < nb; i += NTHR) s_hist[i] = 0;
    __syncthreads();

    for (int i = 0; i < PT; ++i) {
      int e = i * NTHR + tid;                     // strided: LDS bank-conflict free
      unsigned k = f2key(__float_as_uint(s_vals[e]));
      bool ok = true;
      if (p != 0) ok = ((k >> (shift + bits)) == prefHigh);
      if (ok) atomicAdd(&s_hist[(k >> shift) & (nb - 1u)], 1u);
    }
    __syncthreads();

    // group sums (8 bins/group), padded to NTHR with zeros
    const unsigned ngroups = nb >> 3;
    unsigned gsum = 0;
    if ((unsigned)tid < ngroups) {
      #pragma unroll
      for (int j = 0; j < 8; ++j) gsum += s_hist[tid * 8 + j];
    }
    s_suf[tid] = ((unsigned)tid < ngroups) ? gsum : 0;
    __syncthreads();

    // inclusive suffix scan (Hillis-Steele) over NTHR entries
    for (int off = 1; off < NTHR; off <<= 1) {
      unsigned v = s_suf[tid];
      unsigned add = (tid + off < NTHR) ? s_suf[tid + off] : 0;
      __syncthreads();
      s_suf[tid] = v + add;
      __syncthreads();
    }

    // exactly one group straddles the krem-th largest; it refines within its 8 bins
    if ((unsigned)tid < ngroups) {
      unsigned incl  = s_suf[tid];
      unsigned above = (tid + 1 < NTHR) ? s_suf[tid + 1] : 0;
      if (incl >= krem && above < krem) {
        unsigned acc = above;
        for (int b = 7; b >= 0; --b) {
          unsigned c = s_hist[tid * 8 + b];
          if (acc + c >= krem) { s_selbin = (unsigned)tid * 8u + (unsigned)b; s_krem = krem - acc; break; }
          acc += c;
        }
      }
    }
    __syncthreads();
    prefHigh = (prefHigh << bits) | s_selbin;
    krem = s_krem;
    __syncthreads();
  }

  const unsigned T   = prefHigh;  // full 32-bit threshold key (K-th largest)
  const unsigned rem = krem;      // how many key==T elements belong in top-K

  // ---- Deterministic tie-break: take the `rem` lowest-index elements equal to T ----
  for (int i = 0; i < PT; ++i) {
    int e = i * NTHR + tid;
    unsigned k = f2key(__float_as_uint(s_vals[e]));
    if (k == T) {
      unsigned pos = atomicAdd(&s_eqcount, 1u);
      if (pos < EQCAP) s_eqlist[pos] = (unsigned)e;
    }
  }
  __syncthreads();
  unsigned m = s_eqcount; if (m > EQCAP) m = EQCAP;
  for (unsigned i = tid; i < m; i += NTHR) {
    unsigned ei = s_eqlist[i];
    unsigned rank = 0;
    for (unsigned j = 0; j < m; ++j) rank += (s_eqlist[j] < ei) ? 1u : 0u;  // rank by index
    if (rank < rem) atomicOr(&s_bitmap[ei >> 5], 1u << (ei & 31u));
  }
  __syncthreads();

  // ---- Transform row in place: selected ? relu(v) : 0 ----
  for (int i = 0; i < PT; ++i) {
    int e = i * NTHR + tid;
    float f = s_vals[e];
    unsigned k = f2key(__float_as_uint(f));
    bool sel = (k > T) || ((k == T) && ((s_bitmap[e >> 5] >> (e & 31u)) & 1u));
    s_vals[e] = (sel && f > 0.0f) ? f : 0.0f;
  }
  __syncthreads();

  // ---- Async DMA LDS -> global ----
  #pragma unroll
  for (int i = 0; i < PT4; ++i) {
    int e4 = i * NTHR + tid;
    async_l2g_b128(orow + 4 * e4, &s_vals[4 * e4]);
  }
  __builtin_amdgcn_s_wait_asynccnt(0);
}

extern "C" void kernel_launch(void* const* d_in, const int* in_sizes, int n_in,
                              void* d_out, int out_size, void* d_ws, size_t ws_size,
                              hipStream_t stream) {
  (void)n_in; (void)out_size; (void)d_ws; (void)ws_size;
  const float* x = (const float*)d_in[0];
  float* out = (float*)d_out;
  const int B = in_sizes[0] / L_DIM;   // 4096 rows
  topk_relu_scatter_kernel<<<dim3(B), dim3(NTHR), 0, stream>>>(x, out);
}